// Vocoder_8280696947392
// MI455X (gfx1250) — compile-verified
//
#include <hip/hip_runtime.h>
#include <math.h>

#define HIDDEN  512
#define NHARM   64
#define NBANDS  41
#define FRAME   80
#define FSR     16000.0f
#define ATTEN   0.02f
#define TFRAMES 2000
#define BATCH   4
#define TSAMP   (TFRAMES * FRAME)   /* 160000 */
#define RLEN    16001
#define TWO_PI  6.28318530717958647692f
#define INV_2PI 0.15915494309189533577f

typedef float v2f __attribute__((ext_vector_type(2)));
typedef float v8f __attribute__((ext_vector_type(8)));
typedef unsigned int v4u __attribute__((ext_vector_type(4)));
typedef int v4i __attribute__((ext_vector_type(4)));
typedef int v8i __attribute__((ext_vector_type(8)));

#if defined(__has_builtin)
#if __has_builtin(__builtin_amdgcn_tensor_load_to_lds) && __has_builtin(__builtin_amdgcn_s_wait_tensorcnt)
#define HAVE_TDM 1
#endif
#endif

// ---------------------------------------------------------------------------
// 0) Weight transpose for the encoder GEMMs:
//    Wt[layer][kk][ci][co] = w_dil[layer][co][ci][kk]
//    -> A-fragment loads become lane-contiguous (coalesced over co).
// ---------------------------------------------------------------------------
__global__ void k_transpose_w(const float* __restrict__ w,   // [5][512][512][3]
                              float* __restrict__ wt) {      // [5][3][512][512]
    size_t idx = (size_t)blockIdx.x * blockDim.x + threadIdx.x;
    const size_t total = 5ull * HIDDEN * HIDDEN * 3;
    if (idx >= total) return;
    int kk = idx % 3;
    int ci = (idx / 3) % HIDDEN;
    int co = (idx / (3 * HIDDEN)) % HIDDEN;
    int l  = idx / (3ull * HIDDEN * HIDDEN);
    wt[(((size_t)l * 3 + kk) * HIDDEN + ci) * HIDDEN + co] = w[idx];
}

// ---------------------------------------------------------------------------
// 1) Input conv: [B,2,TF] -> relu(conv K=3) -> h [B,512,TF]
// ---------------------------------------------------------------------------
__global__ void k_in_conv(const float* __restrict__ f0,
                          const float* __restrict__ loud,
                          const float* __restrict__ w_in,
                          float* __restrict__ h) {
    int t  = blockIdx.x * blockDim.x + threadIdx.x;
    int co = blockIdx.y;
    int b  = blockIdx.z;
    if (t >= TFRAMES) return;
    const float* w = w_in + co * 2 * 3;
    float acc = 0.f;
    #pragma unroll
    for (int ci = 0; ci < 2; ci++) {
        const float* x = (ci == 0 ? f0 : loud) + (size_t)b * TFRAMES;
        #pragma unroll
        for (int k = 0; k < 3; k++) {
            int tt = t + k - 1;
            float xv = (tt >= 0 && tt < TFRAMES) ? x[tt] : 0.f;
            acc = fmaf(w[ci * 3 + k], xv, acc);
        }
    }
    h[((size_t)b * HIDDEN + co) * TFRAMES + t] = fmaxf(acc, 0.f);
}

// ---------------------------------------------------------------------------
// 2) Dilated conv layer via V_WMMA_F32_16X16X4_F32.
//    hout = hin + relu(conv(hin, W, dil)), 'same' padding (pad = dil).
//    One wave computes a 16(co) x 80(t) strip: 5 C-tiles sharing each A
//    fragment. K-loop: 3 taps x 512 channels in steps of 4.
//    A (16x4): lane<16 -> M=lane, K={0,1}; lane>=16 -> M=lane-16, K={2,3}
//    B (4x16): v0: lanes0-15 K=0 / lanes16-31 K=2; v1: K=1 / K=3
//    C (16x16): vi: lanes0-15 M=i / lanes16-31 M=i+8, N=lane&15
// ---------------------------------------------------------------------------
#define NTILE 5
__global__ void __launch_bounds__(32)
k_dil_conv_wmma(const float* __restrict__ hin,
                const float* __restrict__ wt,    // [3][512][512]  (kk, ci, co)
                float* __restrict__ hout,
                int dil) {
    const int lane = threadIdx.x;
    const int co0  = blockIdx.x * 16;
    const int t0   = blockIdx.y * (16 * NTILE);
    const int b    = blockIdx.z;
    const int hl   = lane >> 4;     // 0: lanes 0-15, 1: lanes 16-31
    const int l15  = lane & 15;
    const int kqb  = hl * 2;        // K-pair base within the 4-wide K slab
    const int acol = co0 + l15;     // A-fragment column (co) for this lane

    v8f c[NTILE];
    #pragma unroll
    for (int j = 0; j < NTILE; j++) c[j] = (v8f){};

    const float* hb = hin + (size_t)b * HIDDEN * TFRAMES;

    for (int kk = 0; kk < 3; kk++) {
        const int tshift = (kk - 1) * dil;
        int  col[NTILE];
        bool cok[NTILE];
        #pragma unroll
        for (int j = 0; j < NTILE; j++) {
            col[j] = t0 + j * 16 + l15 + tshift;
            cok[j] = (col[j] >= 0) && (col[j] < TFRAMES);
        }
        const float* wk = wt + (size_t)kk * HIDDEN * HIDDEN;
        for (int ci0 = 0; ci0 < HIDDEN; ci0 += 4) {
            const float* wr0 = wk + (size_t)(ci0 + kqb)     * HIDDEN;
            const float* wr1 = wk + (size_t)(ci0 + kqb + 1) * HIDDEN;
            __builtin_prefetch(wr0 + 4 * HIDDEN + acol, 0, 0);
            v2f a;
            a.x = wr0[acol];
            a.y = wr1[acol];
            const float* hr0 = hb + (size_t)(ci0 + kqb)     * TFRAMES;
            const float* hr1 = hb + (size_t)(ci0 + kqb + 1) * TFRAMES;
            #pragma unroll
            for (int j = 0; j < NTILE; j++) {
                v2f bm;
                bm.x = cok[j] ? hr0[col[j]] : 0.f;
                bm.y = cok[j] ? hr1[col[j]] : 0.f;
                c[j] = __builtin_amdgcn_wmma_f32_16x16x4_f32(
                           false, a, false, bm, (short)0, c[j], false, false);
            }
        }
    }
    #pragma unroll
    for (int j = 0; j < NTILE; j++) {
        #pragma unroll
        for (int i = 0; i < 8; i++) {
            int row = co0 + i + hl * 8;
            int col = t0 + j * 16 + l15;
            size_t idx = ((size_t)b * HIDDEN + row) * TFRAMES + col;
            hout[idx] = hin[idx] + fmaxf(c[j][i], 0.f);
        }
    }
}

// ---------------------------------------------------------------------------
// 3) 1x1 head conv: out[b,co,t] = dot(h[b,:,t], w[co,:])
// ---------------------------------------------------------------------------
__global__ void k_head(const float* __restrict__ h,
                       const float* __restrict__ w,   // [Co][512]
                       float* __restrict__ out, int Co) {
    int idx = blockIdx.x * blockDim.x + threadIdx.x;
    int total = BATCH * Co * TFRAMES;
    if (idx >= total) return;
    int t  = idx % TFRAMES;
    int co = (idx / TFRAMES) % Co;
    int b  = idx / (TFRAMES * Co);
    const float* hb = h + (size_t)b * HIDDEN * TFRAMES + t;
    const float* wr = w + (size_t)co * HIDDEN;
    float acc = 0.f;
    for (int ci = 0; ci < HIDDEN; ci++)
        acc = fmaf(hb[(size_t)ci * TFRAMES], wr[ci], acc);
    out[idx] = acc;
}

// ---------------------------------------------------------------------------
// 4) softmax over the 64-harmonic axis (in place), one thread per (b,t)
// ---------------------------------------------------------------------------
__global__ void k_softmax64(float* __restrict__ p_) {
    int t = blockIdx.x * blockDim.x + threadIdx.x;
    int b = blockIdx.y;
    if (t >= TFRAMES) return;
    float* p = p_ + (size_t)b * NHARM * TFRAMES + t;
    float mx = -1e30f;
    for (int n = 0; n < NHARM; n++) mx = fmaxf(mx, p[(size_t)n * TFRAMES]);
    float s = 0.f;
    for (int n = 0; n < NHARM; n++) {
        float e = __expf(p[(size_t)n * TFRAMES] - mx);
        p[(size_t)n * TFRAMES] = e;
        s += e;
    }
    float inv = 1.f / s;
    for (int n = 0; n < NHARM; n++) p[(size_t)n * TFRAMES] *= inv;
}

// ---------------------------------------------------------------------------
// 5) mod_sigmoid in place: scale * (2*sigmoid(x)^ln(10) + 1e-7)
// ---------------------------------------------------------------------------
__global__ void k_modsig(float* __restrict__ x, int n, float scale) {
    int i = blockIdx.x * blockDim.x + threadIdx.x;
    if (i >= n) return;
    float s = 1.f / (1.f + __expf(-x[i]));
    x[i] = (2.f * __powf(s, 2.302585092994046f) + 1e-7f) * scale;
}

// ---------------------------------------------------------------------------
// 6) linear upsample of f0 to sample rate (half-pixel centers, edge clamp)
// ---------------------------------------------------------------------------
__global__ void k_f0up(const float* __restrict__ f0, float* __restrict__ f0u) {
    int t = blockIdx.x * blockDim.x + threadIdx.x;
    int b = blockIdx.y;
    if (t >= TSAMP) return;
    float xi = (t + 0.5f) * (1.0f / FRAME) - 0.5f;
    xi = fminf(fmaxf(xi, 0.f), (float)(TFRAMES - 1));
    int i0 = (int)xi;
    int i1 = min(i0 + 1, TFRAMES - 1);
    float fr = xi - (float)i0;
    const float* p = f0 + (size_t)b * TFRAMES;
    f0u[(size_t)b * TSAMP + t] = p[i0] + fr * (p[i1] - p[i0]);
}

// ---------------------------------------------------------------------------
// 7) phase = 2*pi*cumsum(f0_up)/FS  — per-batch block scan with carry
// ---------------------------------------------------------------------------
__global__ void __launch_bounds__(256)
k_scan_phase(const float* __restrict__ f0u, float* __restrict__ phase) {
    int b = blockIdx.x;
    __shared__ float sh[256];
    __shared__ float carry;
    if (threadIdx.x == 0) carry = 0.f;
    __syncthreads();
    const float* x = f0u + (size_t)b * TSAMP;
    float* p = phase + (size_t)b * TSAMP;
    for (int base = 0; base < TSAMP; base += 256) {
        int i = base + threadIdx.x;
        sh[threadIdx.x] = x[i];
        __syncthreads();
        for (int off = 1; off < 256; off <<= 1) {
            float tv = (threadIdx.x >= off) ? sh[threadIdx.x - off] : 0.f;
            __syncthreads();
            sh[threadIdx.x] += tv;
            __syncthreads();
        }
        float cseen = carry;
        p[i] = (TWO_PI / FSR) * (cseen + sh[threadIdx.x]);
        __syncthreads();
        if (threadIdx.x == 255) carry = cseen + sh[255];
        __syncthreads();
    }
}

// ---------------------------------------------------------------------------
// 8) harmonic oscillator
// ---------------------------------------------------------------------------
__global__ void k_osc(const float* __restrict__ phase,
                      const float* __restrict__ f0u,
                      const float* __restrict__ cn_sin,
                      const float* __restrict__ cn_cos,
                      const float* __restrict__ an_sin,
                      const float* __restrict__ an_cos,
                      float* __restrict__ audio) {
    int t = blockIdx.x * blockDim.x + threadIdx.x;
    int b = blockIdx.y;
    if (t >= TSAMP) return;
    float ph = phase[(size_t)b * TSAMP + t];
    ph -= TWO_PI * floorf(ph * INV_2PI);        // wrap (valid mod 2*pi, n integer)
    float f0 = f0u[(size_t)b * TSAMP + t];

    float xi = (t + 0.5f) * (1.0f / FRAME) - 0.5f;
    xi = fminf(fmaxf(xi, 0.f), (float)(TFRAMES - 1));
    int i0 = (int)xi;
    int i1 = min(i0 + 1, TFRAMES - 1);
    float fr = xi - (float)i0;

    const float* as_p = an_sin + (size_t)b * TFRAMES;
    const float* ac_p = an_cos + (size_t)b * TFRAMES;
    float as = as_p[i0] + fr * (as_p[i1] - as_p[i0]);
    float ac = ac_p[i0] + fr * (ac_p[i1] - ac_p[i0]);

    const float* cs = cn_sin + (size_t)b * NHARM * TFRAMES;
    const float* cc = cn_cos + (size_t)b * NHARM * TFRAMES;

    float sum_s = 0.f, sum_c = 0.f;
    for (int n = 1; n <= NHARM; n++) {
        if ((float)n * f0 < 0.5f * FSR) {
            float a = (float)n * ph;
            a -= TWO_PI * floorf(a * INV_2PI);
            float sv, cv;
            __sincosf(a, &sv, &cv);
            const float* rs = cs + (size_t)(n - 1) * TFRAMES;
            const float* rc = cc + (size_t)(n - 1) * TFRAMES;
            float csv = rs[i0] + fr * (rs[i1] - rs[i0]);
            float ccv = rc[i0] + fr * (rc[i1] - rc[i0]);
            sum_s = fmaf(csv, sv, sum_s);
            sum_c = fmaf(ccv, cv, sum_c);
        }
    }
    audio[(size_t)b * TSAMP + t] = as * sum_s + ac * sum_c;
}

// ---------------------------------------------------------------------------
// 9) filtered noise: per-frame 41-bin rDFT, real gain H, irDFT; add to audio
// ---------------------------------------------------------------------------
__global__ void __launch_bounds__(128)
k_noise(const float* __restrict__ noise,
        const float* __restrict__ Hh,          // [B][41][TF]
        float* __restrict__ audio) {
    int f = blockIdx.x;
    int b = blockIdx.y;
    __shared__ float sn[FRAME];
    __shared__ float sre[NBANDS];
    __shared__ float sim[NBANDS];
    int tid = threadIdx.x;
    if (tid < FRAME) sn[tid] = noise[((size_t)b * TFRAMES + f) * FRAME + tid];
    __syncthreads();
    if (tid < NBANDS) {
        float re = 0.f, im = 0.f;
        for (int k = 0; k < FRAME; k++) {
            float ang = -TWO_PI * (float)(tid * k) * (1.0f / FRAME);
            float sv, cv;
            __sincosf(ang, &sv, &cv);
            re = fmaf(sn[k], cv, re);
            im = fmaf(sn[k], sv, im);
        }
        float g = Hh[((size_t)b * NBANDS + tid) * TFRAMES + f];
        sre[tid] = re * g;
        sim[tid] = im * g;
    }
    __syncthreads();
    if (tid < FRAME) {
        float acc = sre[0];
        for (int m = 1; m < NBANDS - 1; m++) {
            float ang = TWO_PI * (float)(m * tid) * (1.0f / FRAME);
            float sv, cv;
            __sincosf(ang, &sv, &cv);
            acc += 2.f * (sre[m] * cv - sim[m] * sv);
        }
        acc += sre[NBANDS - 1] * ((tid & 1) ? -1.f : 1.f);   // Nyquist bin
        size_t idx = (size_t)b * TSAMP + (size_t)f * FRAME + tid;
        audio[idx] += acc * (1.0f / FRAME);
    }
}

// ---------------------------------------------------------------------------
// 10) 16001-tap FIR reverb ('same'): y[t] = sum_k w[k]*x[t+k-8000]
//     Weight chunks staged into LDS by the Tensor Data Mover (one wave
//     issues a 1-D D#; TDM OOB reads return zero, so the tail chunk is
//     zero-padded for free). Audio window staged by regular VMEM.
// ---------------------------------------------------------------------------
#define RCHUNK 256
__global__ void __launch_bounds__(256)
k_reverb(const float* __restrict__ audio,
         const float* __restrict__ w,
         float* __restrict__ out) {
    int t0  = blockIdx.x * RCHUNK;
    int b   = blockIdx.y;
    int tid = threadIdx.x;
    __shared__ float sw[RCHUNK];
    __shared__ float sx[2 * RCHUNK];
    const float* x = audio + (size_t)b * TSAMP;
    float acc = 0.f;
    for (int k0 = 0; k0 < RLEN; k0 += RCHUNK) {
        int base = t0 + k0 - (RLEN - 1) / 2;
        int i1 = base + tid;
        sx[tid] = (i1 >= 0 && i1 < TSAMP) ? x[i1] : 0.f;
        int i2 = base + RCHUNK + tid;
        sx[RCHUNK + tid] = (i2 >= 0 && i2 < TSAMP) ? x[i2] : 0.f;
        __builtin_prefetch(x + base + 2 * RCHUNK + tid, 0, 0);
#ifdef HAVE_TDM
        if (tid < 32) {
            // ---- Tensor DMA descriptor: 1-D tile of RCHUNK f32 taps ----
            unsigned lds_off = (unsigned)(size_t)(void*)sw;          // LDS aperture: low 32 bits = LDS offset
            unsigned long long ga = (unsigned long long)(size_t)(w + k0);
            unsigned td0 = (unsigned)(RLEN - k0);                    // remaining taps; OOB reads -> 0
            v4u g0 = {};
            g0[0] = 1u;                                              // count=1, user mode
            g0[1] = lds_off;                                         // lds_addr
            g0[2] = (unsigned)ga;                                    // global_addr[31:0]
            g0[3] = (unsigned)((ga >> 32) & 0x1FFFFFFull) | (2u << 30); // global_addr[56:32] | type=2
            v8i g1 = {};
            g1[0] = (2 << 16);                                       // data_size = 4 bytes
            g1[1] = (int)((td0 & 0xFFFFu) << 16);                    // tensor_dim0[15:0]  (bits 63:48)
            g1[2] = (int)((td0 >> 16) & 0xFFFFu) | (1 << 16);        // tensor_dim0[31:16] | tensor_dim1=1
            g1[3] = (int)(RCHUNK << 16);                             // tile_dim0 = 256 (bits 127:112)
            g1[4] = 0;                                               // tile_dim1/2 unused
            g1[5] = (int)td0;                                        // tensor_dim0_stride[31:0]
            g1[6] = 0;
            g1[7] = 0;
            v4i g2 = {};
            v4i g3 = {};
#if defined(__clang_major__) && (__clang_major__ >= 23)
            v8i g4 = {};
            __builtin_amdgcn_tensor_load_to_lds(g0, g1, g2, g3, g4, 0);
#else
            __builtin_amdgcn_tensor_load_to_lds(g0, g1, g2, g3, 0);
#endif
            __builtin_amdgcn_s_wait_tensorcnt(0);
        }
#else
        {
            int kw = k0 + tid;
            sw[tid] = (kw < RLEN) ? w[kw] : 0.f;
        }
#endif
        __syncthreads();
        #pragma unroll 8
        for (int kk = 0; kk < RCHUNK; kk++)
            acc = fmaf(sw[kk], sx[tid + kk], acc);
        __syncthreads();
    }
    out[(size_t)b * TSAMP + t0 + tid] = acc;
}

// ---------------------------------------------------------------------------
// launch
// ---------------------------------------------------------------------------
extern "C" void kernel_launch(void* const* d_in, const int* in_sizes, int n_in,
                              void* d_out, int out_size, void* d_ws, size_t ws_size,
                              hipStream_t stream) {
    const float* f0    = (const float*)d_in[0];
    const float* loud  = (const float*)d_in[1];
    const float* noise = (const float*)d_in[2];
    const float* w_in  = (const float*)d_in[3];
    const float* w_dil = (const float*)d_in[4];
    const float* w_cs  = (const float*)d_in[5];
    const float* w_cc  = (const float*)d_in[6];
    const float* w_as  = (const float*)d_in[7];
    const float* w_ac  = (const float*)d_in[8];
    const float* w_h   = (const float*)d_in[9];
    const float* w_rev = (const float*)d_in[10];
    float* out = (float*)d_out;

    float* ws = (float*)d_ws;
    const size_t HSZ = (size_t)BATCH * HIDDEN * TFRAMES;   // 4,096,000
    float* h0     = ws;                    ws += HSZ;
    float* h1     = ws;                    ws += HSZ;
    float* wt     = ws;                    ws += 5ull * 3 * HIDDEN * HIDDEN;
    float* cn_sin = ws;                    ws += (size_t)BATCH * NHARM * TFRAMES;
    float* cn_cos = ws;                    ws += (size_t)BATCH * NHARM * TFRAMES;
    float* an_sin = ws;                    ws += (size_t)BATCH * TFRAMES;
    float* an_cos = ws;                    ws += (size_t)BATCH * TFRAMES;
    float* Hh     = ws;                    ws += (size_t)BATCH * NBANDS * TFRAMES;
    float* f0u    = ws;                    ws += (size_t)BATCH * TSAMP;
    float* phase  = ws;                    ws += (size_t)BATCH * TSAMP;
    float* audio  = ws;                    ws += (size_t)BATCH * TSAMP;

    // 0) weight transpose for encoder GEMMs
    {
        size_t tot = 5ull * HIDDEN * HIDDEN * 3;
        k_transpose_w<<<(unsigned)((tot + 255) / 256), 256, 0, stream>>>(w_dil, wt);
    }
    // 1) input conv
    {
        dim3 g((TFRAMES + 255) / 256, HIDDEN, BATCH);
        k_in_conv<<<g, 256, 0, stream>>>(f0, loud, w_in, h0);
    }
    // 2) 5 dilated conv layers (f32 WMMA), ping-pong h0/h1
    const int dils[5] = {1, 2, 4, 8, 16};
    float* cur = h0;
    float* nxt = h1;
    for (int i = 0; i < 5; i++) {
        dim3 g(HIDDEN / 16, TFRAMES / (16 * NTILE), BATCH);   // 32 x 25 x 4, exact
        k_dil_conv_wmma<<<g, 32, 0, stream>>>(
            cur, wt + (size_t)i * 3 * HIDDEN * HIDDEN, nxt, dils[i]);
        float* tmp = cur; cur = nxt; nxt = tmp;
    }
    // 3) heads
    {
        int tot = BATCH * NHARM * TFRAMES;
        k_head<<<(tot + 255) / 256, 256, 0, stream>>>(cur, w_cs, cn_sin, NHARM);
        k_head<<<(tot + 255) / 256, 256, 0, stream>>>(cur, w_cc, cn_cos, NHARM);
        int tot1 = BATCH * TFRAMES;
        k_head<<<(tot1 + 255) / 256, 256, 0, stream>>>(cur, w_as, an_sin, 1);
        k_head<<<(tot1 + 255) / 256, 256, 0, stream>>>(cur, w_ac, an_cos, 1);
        int totH = BATCH * NBANDS * TFRAMES;
        k_head<<<(totH + 255) / 256, 256, 0, stream>>>(cur, w_h, Hh, NBANDS);
    }
    // 4) activations
    {
        dim3 g((TFRAMES + 255) / 256, BATCH);
        k_softmax64<<<g, 256, 0, stream>>>(cn_sin);
        k_softmax64<<<g, 256, 0, stream>>>(cn_cos);
        int n1 = BATCH * TFRAMES;
        k_modsig<<<(n1 + 255) / 256, 256, 0, stream>>>(an_sin, n1, 1.0f);
        k_modsig<<<(n1 + 255) / 256, 256, 0, stream>>>(an_cos, n1, 1.0f);
        int nH = BATCH * NBANDS * TFRAMES;
        k_modsig<<<(nH + 255) / 256, 256, 0, stream>>>(Hh, nH, ATTEN);
    }
    // 5) oscillator path
    {
        dim3 g((TSAMP + 255) / 256, BATCH);
        k_f0up<<<g, 256, 0, stream>>>(f0, f0u);
        k_scan_phase<<<BATCH, 256, 0, stream>>>(f0u, phase);
        k_osc<<<g, 256, 0, stream>>>(phase, f0u, cn_sin, cn_cos, an_sin, an_cos, audio);
    }
    // 6) filtered noise added to audio
    {
        dim3 g(TFRAMES, BATCH);
        k_noise<<<g, 128, 0, stream>>>(noise, Hh, audio);
    }
    // 7) reverb -> output
    {
        dim3 g(TSAMP / RCHUNK, BATCH);
        k_reverb<<<g, RCHUNK, 0, stream>>>(audio, w_rev, out);
    }
    (void)in_sizes; (void)n_in; (void)out_size; (void)ws_size;
}